// SGC_61795989455225
// MI455X (gfx1250) — compile-verified
//
#include <hip/hip_runtime.h>

#define N_NODES   100000
#define N_EDGES   3200000
#define N_FEAT    128
#define N_CLASSES 40
#define MTILES    (N_NODES / 16)   // 6250
#define NTILES    3                // ceil(40/16)
#define WAVES_PER_BLOCK 8

typedef __attribute__((ext_vector_type(2))) float v2f;
typedef __attribute__((ext_vector_type(8))) float v8f;

// ---------------- degree / normalization ----------------

__global__ void deg_init_kernel(float* __restrict__ deg) {
    int i = blockIdx.x * blockDim.x + threadIdx.x;
    if (i < N_NODES) deg[i] = 1.0f;   // self-loop contribution
}

__global__ void deg_count_kernel(const int* __restrict__ dst, float* __restrict__ deg) {
    int e = blockIdx.x * blockDim.x + threadIdx.x;
    if (e < N_EDGES) atomicAdd(&deg[dst[e]], 1.0f);
}

__global__ void deg_to_dinv_kernel(float* __restrict__ deg) {
    int i = blockIdx.x * blockDim.x + threadIdx.x;
    if (i < N_NODES) deg[i] = rsqrtf(deg[i]);  // deg > 0 always (self-loops)
}

// ---------------- propagation hop ----------------
// y[i,:] = x[i,:] * dinv[i]^2   (self-loop term). One wave32 per node, float4/lane.
__global__ void hop_init_kernel(const float* __restrict__ x,
                                const float* __restrict__ dinv,
                                float* __restrict__ y) {
    int t    = blockIdx.x * blockDim.x + threadIdx.x;
    int node = t >> 5;
    int lane = t & 31;
    if (node < N_NODES) {
        float d = dinv[node];
        float w = d * d;
        float4 v = ((const float4*)(x + (long long)node * N_FEAT))[lane];
        float4 o; o.x = v.x * w; o.y = v.y * w; o.z = v.z * w; o.w = v.w * w;
        ((float4*)(y + (long long)node * N_FEAT))[lane] = o;
    }
}

// y[dst,:] += x[src,:] * dinv[src]*dinv[dst].  One wave32 per edge; lane handles
// a float4 chunk (32 * 16B = 512B = one node's feature row). x fits in 192MB L2,
// so gathers + f32 atomics stay on-chip.
__global__ void hop_edges_kernel(const float* __restrict__ xin,
                                 const int* __restrict__ src,
                                 const int* __restrict__ dst,
                                 const float* __restrict__ dinv,
                                 float* __restrict__ y) {
    int t    = blockIdx.x * blockDim.x + threadIdx.x;
    int e    = t >> 5;
    int lane = t & 31;
    if (e < N_EDGES) {
        int s = src[e];
        int d = dst[e];
        float w = dinv[s] * dinv[d];
        float4 v = ((const float4*)(xin + (long long)s * N_FEAT))[lane];
        float* yo = y + (long long)d * N_FEAT + lane * 4;
        atomicAdd(yo + 0, v.x * w);
        atomicAdd(yo + 1, v.y * w);
        atomicAdd(yo + 2, v.z * w);
        atomicAdd(yo + 3, v.w * w);
    }
}

// ---------------- final linear: out = X @ W^T + b via V_WMMA_F32_16X16X4_F32 ----
// One wave32 per 16x16 output tile. K = 128 in steps of 4 -> 32 WMMAs.
// A (16x4 f32): lanes 0-15 hold K={k,k+1}, lanes 16-31 hold K={k+2,k+3} (ISA 7.12.2).
// B (4x16 f32): mirrored layout with N across lanes. B[k][n] = W[n][k].
// D (16x16 f32): VGPR j -> rows {j, j+8}; lanes 0-15 N=0..15, lanes 16-31 N=0..15.
__global__ void __launch_bounds__(256)
sgc_linear_wmma_kernel(const float* __restrict__ X,
                       const float* __restrict__ Wm,
                       const float* __restrict__ bias,
                       float* __restrict__ out) {
    const int wave = threadIdx.x >> 5;
    const int lane = threadIdx.x & 31;
    const int tile = blockIdx.x * WAVES_PER_BLOCK + wave;
    if (tile >= MTILES * NTILES) return;     // wave-uniform exit

    const int mtile = tile / NTILES;
    const int ntile = tile % NTILES;
    const int m0 = mtile * 16;
    const int n0 = ntile * 16;

    const int half = lane >> 4;     // 0: K+{0,1}, 1: K+{2,3}
    const int l    = lane & 15;
    const int koff = half * 2;

    const int arow = m0 + l;                        // A row (always < N_NODES)
    const int ncol = n0 + l;                        // B column (class)
    const int nclamp = (ncol < N_CLASSES) ? ncol : (N_CLASSES - 1);
    const float nmask = (ncol < N_CLASSES) ? 1.0f : 0.0f;  // zero-pad classes 40..47

    const float* aptr = X  + (long long)arow   * N_FEAT + koff;
    const float* bptr = Wm + (long long)nclamp * N_FEAT + koff;

    v8f c = {0.0f, 0.0f, 0.0f, 0.0f, 0.0f, 0.0f, 0.0f, 0.0f};

#pragma unroll
    for (int k = 0; k < N_FEAT; k += 4) {
        float2 av = *(const float2*)(aptr + k);   // 8B aligned: koff even, k % 4 == 0
        float2 bv = *(const float2*)(bptr + k);
        v2f a; a.x = av.x;          a.y = av.y;
        v2f b; b.x = bv.x * nmask;  b.y = bv.y * nmask;
        // 8 args: (neg_a, A, neg_b, B, c_mod, C, reuse_a, reuse_b)
        c = __builtin_amdgcn_wmma_f32_16x16x4_f32(
                false, a, false, b, (short)0, c, false, false);
    }

    const float bv = (ncol < N_CLASSES) ? bias[ncol] : 0.0f;
#pragma unroll
    for (int j = 0; j < 8; ++j) {
        int m = m0 + j + half * 8;
        if (ncol < N_CLASSES)
            out[(long long)m * N_CLASSES + ncol] = c[j] + bv;
    }
}

// ---------------- launch ----------------

extern "C" void kernel_launch(void* const* d_in, const int* in_sizes, int n_in,
                              void* d_out, int out_size, void* d_ws, size_t ws_size,
                              hipStream_t stream) {
    const float* x    = (const float*)d_in[0];   // [N_NODES, N_FEAT]
    const float* Wm   = (const float*)d_in[1];   // [N_CLASSES, N_FEAT]
    const float* bias = (const float*)d_in[2];   // [N_CLASSES]
    const int*   ei   = (const int*)d_in[3];     // [2, N_EDGES]
    const int* src = ei;
    const int* dst = ei + N_EDGES;

    // Workspace layout: dinv (400KB) | y1 (51.2MB) | y2 (51.2MB)
    char* ws = (char*)d_ws;
    float* dinv = (float*)ws;
    size_t off = ((size_t)N_NODES * sizeof(float) + 4095) & ~(size_t)4095;
    float* y1 = (float*)(ws + off);
    float* y2 = (float*)(ws + off + (size_t)N_NODES * N_FEAT * sizeof(float));
    float* out = (float*)d_out;

    const int B = 256;

    // Degrees -> dinv (fully re-initialized every call; ws is not re-poisoned)
    deg_init_kernel   <<<(N_NODES + B - 1) / B, B, 0, stream>>>(dinv);
    deg_count_kernel  <<<(N_EDGES + B - 1) / B, B, 0, stream>>>(dst, dinv);
    deg_to_dinv_kernel<<<(N_NODES + B - 1) / B, B, 0, stream>>>(dinv);

    // Hop 1: y1 = A_hat * x
    {
        int t = N_NODES * 32;
        hop_init_kernel <<<(t + B - 1) / B, B, 0, stream>>>(x, dinv, y1);
        long long te = (long long)N_EDGES * 32;
        hop_edges_kernel<<<(unsigned)((te + B - 1) / B), B, 0, stream>>>(x, src, dst, dinv, y1);
    }
    // Hop 2: y2 = A_hat * y1
    {
        int t = N_NODES * 32;
        hop_init_kernel <<<(t + B - 1) / B, B, 0, stream>>>(y1, dinv, y2);
        long long te = (long long)N_EDGES * 32;
        hop_edges_kernel<<<(unsigned)((te + B - 1) / B), B, 0, stream>>>(y1, src, dst, dinv, y2);
    }

    // Final linear with f32 WMMA
    int ntiles = MTILES * NTILES;
    sgc_linear_wmma_kernel<<<(ntiles + WAVES_PER_BLOCK - 1) / WAVES_PER_BLOCK, B, 0, stream>>>(
        y2, Wm, bias, out);
}